// MultiheadAttention2D_66151086293482
// MI455X (gfx1250) — compile-verified
//
#include <hip/hip_runtime.h>
#include <hip/hip_bf16.h>

typedef __attribute__((ext_vector_type(2))) float v2f;
typedef __attribute__((ext_vector_type(8))) float v8f;

__device__ __forceinline__ v8f wmma_f32(v2f a, v2f b, v8f c) {
  // V_WMMA_F32_16X16X4_F32: D = A(16x4) * B(4x16) + C(16x16)
  return __builtin_amdgcn_wmma_f32_16x16x4_f32(
      /*neg_a=*/false, a, /*neg_b=*/false, b,
      /*c_mod=*/(short)0, c, /*reuse_a=*/false, /*reuse_b=*/false);
}

// ---------------------------------------------------------------------------
// Generic GEMM: C[M,512] = (A[M,512] @ W[512,512] + bias[512]) * alpha
// 256 threads = 8 waves arranged 4x2; each wave computes a 32x32 tile
// (2x2 WMMA 16x16 frags). Block tile = 128x64. Grid = (512/64, M/128).
// ---------------------------------------------------------------------------
__global__ __launch_bounds__(256) void gemm_bias_kernel(
    const float* __restrict__ A, const float* __restrict__ W,
    const float* __restrict__ bias, float* __restrict__ C,
    int M, float alpha)
{
  const int K = 512, N = 512;
  const int lane = threadIdx.x & 31;
  const int wave = threadIdx.x >> 5;
  const int wm = wave >> 1, wn = wave & 1;
  const int rowBase = blockIdx.y * 128 + wm * 32;
  const int colBase = blockIdx.x * 64 + wn * 32;
  const int ml = lane & 15;          // M (A) / N (B,C) index within 16
  const int hi = lane >> 4;
  const int kk = hi * 2;             // K sub-offset for this half-wave

  v8f c00 = {}, c01 = {}, c10 = {}, c11 = {};

  const float* a0p = A + (size_t)(rowBase + ml) * K + kk;
  const float* a1p = A + (size_t)(rowBase + 16 + ml) * K + kk;
  const float* b0p = W + (size_t)kk * N + colBase + ml;
  const float* b1p = b0p + 16;

  #pragma unroll 4
  for (int k0 = 0; k0 < K; k0 += 4) {
    v2f a0 = *(const v2f*)(a0p + k0);
    v2f a1 = *(const v2f*)(a1p + k0);
    v2f b0, b1;
    b0.x = b0p[(size_t)k0 * N];
    b0.y = b0p[(size_t)k0 * N + N];
    b1.x = b1p[(size_t)k0 * N];
    b1.y = b1p[(size_t)k0 * N + N];
    c00 = wmma_f32(a0, b0, c00);
    c01 = wmma_f32(a0, b1, c01);
    c10 = wmma_f32(a1, b0, c10);
    c11 = wmma_f32(a1, b1, c11);
  }

  #pragma unroll
  for (int r = 0; r < 8; ++r) {
    int row0 = rowBase + r + 8 * hi;
    int col0 = colBase + ml;
    C[(size_t)row0 * N + col0]             = (c00[r] + bias[col0])      * alpha;
    C[(size_t)row0 * N + col0 + 16]        = (c01[r] + bias[col0 + 16]) * alpha;
    C[(size_t)(row0 + 16) * N + col0]      = (c10[r] + bias[col0])      * alpha;
    C[(size_t)(row0 + 16) * N + col0 + 16] = (c11[r] + bias[col0 + 16]) * alpha;
  }
}

// ---------------------------------------------------------------------------
// Flash-style attention. One wave32 owns 16 query rows of one (b, head):
//   S = q @ k^T  (+ pos_logits + xor-mask), online softmax, O += P @ V.
// 128 threads = 4 independent waves per block; grid = 1024 blocks
//  -> 4096 waves = B(8) * NH(8) * (1024/16).
// ---------------------------------------------------------------------------
#define HW     1024
#define DMODEL 512
#define DH     64

__global__ __launch_bounds__(128) void attention_kernel(
    const float* __restrict__ qp, const float* __restrict__ kp,
    const float* __restrict__ vp,
    const unsigned char* __restrict__ qmask,
    const unsigned char* __restrict__ kmask,
    const float* __restrict__ pos, float* __restrict__ outbuf)
{
  __shared__ float lds[4 * 16 * 17];         // per-wave 16x16 P tile, padded
  const int lane = threadIdx.x & 31;
  const int wave = threadIdx.x >> 5;
  float* pbuf = &lds[wave * 16 * 17];

  const int gw    = blockIdx.x * 4 + wave;   // 0..4095
  const int qtile = gw & 63;
  const int bh    = gw >> 6;
  const int b     = bh >> 3;
  const int nh    = bh & 7;
  const int qBase = qtile * 16;

  const int ml = lane & 15;
  const int hi = lane >> 4;
  const int kk = hi * 2;

  // Preload 16x64 q tile as 16 A-fragments (K-steps of 4 over dh=64).
  const float* qrow = qp + (size_t)(b * HW + qBase + ml) * DMODEL + nh * DH + kk;
  v2f aq[16];
  #pragma unroll
  for (int s = 0; s < 16; ++s) aq[s] = *(const v2f*)(qrow + s * 4);

  // q-mask bits for the 8 rows this lane's C-registers cover.
  unsigned int qm[8];
  #pragma unroll
  for (int r = 0; r < 8; ++r)
    qm[r] = qmask[b * HW + qBase + r + 8 * hi];

  float mrun[8], lrun[8];
  #pragma unroll
  for (int r = 0; r < 8; ++r) { mrun[r] = -3.0e38f; lrun[r] = 0.0f; }
  v8f o[4] = {};

  const float* kbase  = kp + (size_t)b * HW * DMODEL + nh * DH;
  const float* vbase  = vp + (size_t)b * HW * DMODEL + nh * DH;
  const float* posrow = pos + (size_t)b * HW * HW;

  for (int kt = 0; kt < 64; ++kt) {
    const int keyBase = kt * 16;

    // ---- S = q @ k^T (16x16 tile, 16 WMMA of K=4) ----
    v8f s = {};
    const float* krow = kbase + (size_t)(keyBase + ml) * DMODEL + kk;
    #pragma unroll
    for (int st = 0; st < 16; ++st) {
      v2f bf = *(const v2f*)(krow + st * 4);
      s = wmma_f32(aq[st], bf, s);
    }

    // ---- add pos_logits + mask, online softmax over this key tile ----
    const int keyCol = keyBase + ml;
    const unsigned int km = kmask[b * HW + keyCol];
    float scl[8], p[8];
    #pragma unroll
    for (int r = 0; r < 8; ++r) {
      const int qRow = qBase + r + 8 * hi;
      float lg = s[r] + posrow[(size_t)qRow * HW + keyCol];
      if ((qm[r] ^ km) != 0u) lg = -3.0e38f;   // additive -inf (finite, NaN-safe)
      float rm = lg;
      rm = fmaxf(rm, __shfl_xor(rm, 1, 32));
      rm = fmaxf(rm, __shfl_xor(rm, 2, 32));
      rm = fmaxf(rm, __shfl_xor(rm, 4, 32));
      rm = fmaxf(rm, __shfl_xor(rm, 8, 32));
      const float mnew = fmaxf(mrun[r], rm);
      const float sc   = __expf(mrun[r] - mnew);
      const float pv   = __expf(lg - mnew);
      float rs = pv;
      rs += __shfl_xor(rs, 1, 32);
      rs += __shfl_xor(rs, 2, 32);
      rs += __shfl_xor(rs, 4, 32);
      rs += __shfl_xor(rs, 8, 32);
      lrun[r] = lrun[r] * sc + rs;
      mrun[r] = mnew;
      scl[r] = sc;
      p[r] = pv;
    }

    // ---- rescale running output ----
    #pragma unroll
    for (int t = 0; t < 4; ++t)
      #pragma unroll
      for (int r = 0; r < 8; ++r) o[t][r] *= scl[r];

    // ---- C-layout P -> LDS -> A-layout fragments (same-wave DS is in-order) --
    #pragma unroll
    for (int r = 0; r < 8; ++r)
      pbuf[(r + 8 * hi) * 17 + ml] = p[r];

    // ---- O += P @ V  (4 K-steps x 4 dv tiles = 16 WMMA) ----
    #pragma unroll
    for (int ks = 0; ks < 4; ++ks) {
      v2f ap;
      ap.x = pbuf[ml * 17 + ks * 4 + kk];
      ap.y = pbuf[ml * 17 + ks * 4 + kk + 1];
      const float* vrow = vbase + (size_t)(keyBase + ks * 4 + kk) * DMODEL;
      #pragma unroll
      for (int t = 0; t < 4; ++t) {
        v2f bf;
        bf.x = vrow[t * 16 + ml];
        bf.y = vrow[DMODEL + t * 16 + ml];
        o[t] = wmma_f32(ap, bf, o[t]);
      }
    }
  }

  // ---- finalize: divide by softmax denom, store [b*1024+q, nh*64+dv] ----
  #pragma unroll
  for (int r = 0; r < 8; ++r) {
    const float inv = 1.0f / lrun[r];
    float* orow = outbuf + (size_t)(b * HW + qBase + r + 8 * hi) * DMODEL + nh * DH;
    #pragma unroll
    for (int t = 0; t < 4; ++t)
      orow[t * 16 + ml] = o[t][r] * inv;
  }
}

// ---------------------------------------------------------------------------
extern "C" void kernel_launch(void* const* d_in, const int* in_sizes, int n_in,
                              void* d_out, int out_size, void* d_ws, size_t ws_size,
                              hipStream_t stream) {
  const float*         q   = (const float*)d_in[0];
  const float*         k   = (const float*)d_in[1];
  const unsigned char* qm  = (const unsigned char*)d_in[2];
  const unsigned char* km  = (const unsigned char*)d_in[3];
  const float*         pos = (const float*)d_in[4];
  const float*         Wq  = (const float*)d_in[5];
  const float*         bq  = (const float*)d_in[6];
  const float*         Wk  = (const float*)d_in[7];
  const float*         bk  = (const float*)d_in[8];
  const float*         Wv  = (const float*)d_in[9];
  const float*         bv  = (const float*)d_in[10];
  const float*         Wo  = (const float*)d_in[11];
  const float*         bo  = (const float*)d_in[12];
  float* out = (float*)d_out;

  const int M = 8192;                 // B*H*W
  const size_t mat = (size_t)M * 512;
  float* qp = (float*)d_ws;           // 16 MB each
  float* kp = qp + mat;
  float* vp = kp + mat;
  float* ab = vp + mat;

  dim3 gg(512 / 64, M / 128);         // (8, 64)
  gemm_bias_kernel<<<gg, 256, 0, stream>>>(q,  Wq, bq, qp, M, 0.125f); // dh^-0.5
  gemm_bias_kernel<<<gg, 256, 0, stream>>>(k,  Wk, bk, kp, M, 1.0f);
  gemm_bias_kernel<<<gg, 256, 0, stream>>>(kp, Wv, bv, vp, M, 1.0f);  // v = Wv(kp)
  attention_kernel<<<1024, 128, 0, stream>>>(qp, kp, vp, qm, km, pos, ab);
  gemm_bias_kernel<<<gg, 256, 0, stream>>>(ab, Wo, bo, out, M, 1.0f);
}